// MeanAggregator_83872121356301
// MI455X (gfx1250) — compile-verified
//
#include <hip/hip_runtime.h>
#include <math.h>

// ---------------------------------------------------------------------------
// MeanAggregator for MI455X (gfx1250, wave32):
//   K1: h = features @ W^T + b  via V_WMMA_F32_16X16X4_F32, fused partial
//       column sum / sum-of-squares for BatchNorm stats.
//   K2: finalize per-column mu / rstd.
//   K3: dedup 17 gather indices per output row, fused BN+tanh, mean-aggregate.
// ---------------------------------------------------------------------------

typedef __attribute__((ext_vector_type(2))) float v2f;
typedef __attribute__((ext_vector_type(8))) float v8f;

#define U_ROWS 50000
#define F_DIM  256
#define E_DIM  128
#define NSAMP  17
#define BN_EPS 1e-5f

// ---------------------------------------------------------------------------
// Kernel 1: 256 threads = 8 waves. Wave w computes the 16x16 tile
// [mBase..mBase+15] x [16w..16w+15] of h. Block covers 16 rows x 128 cols.
// grid.x = U/16 = 3125 (exact).
//
// fp32 WMMA fragment layout (ISA 7.12.2, 32-bit A 16x4):
//   lanes 0-15 : M = lane,    v0 -> K=0, v1 -> K=1
//   lanes 16-31: M = lane-16, v0 -> K=2, v1 -> K=3
// B (4x16) mirrors with N across lanes. C/D: VGPR i holds M = i + 8*half.
// ---------------------------------------------------------------------------
__global__ __launch_bounds__(256)
void gemm_bias_stats(const float* __restrict__ feat, const float* __restrict__ W,
                     const float* __restrict__ bias, float* __restrict__ h,
                     float* __restrict__ psum, float* __restrict__ psq)
{
    const int wave  = threadIdx.x >> 5;
    const int lane  = threadIdx.x & 31;
    const int half  = lane >> 4;       // 0: K pair {k,k+1}, 1: {k+2,k+3}
    const int l15   = lane & 15;
    const int mBase = blockIdx.x * 16;
    const int nBase = wave * 16;

    // A[m,k] = feat[mBase+m, k];  B[k,n] = W[nBase+n, k]  (h = feat @ W^T)
    const float* __restrict__ Arow = feat + (size_t)(mBase + l15) * F_DIM + half * 2;
    const float* __restrict__ Brow = W    + (size_t)(nBase + l15) * F_DIM + half * 2;

    v8f c = {};
    #pragma unroll 8
    for (int k = 0; k < F_DIM; k += 4) {
        v2f a, b2;
        a.x  = Arow[k];  a.y  = Arow[k + 1];
        b2.x = Brow[k];  b2.y = Brow[k + 1];
        // D = A(16x4) x B(4x16) + C ; emits v_wmma_f32_16x16x4_f32
        c = __builtin_amdgcn_wmma_f32_16x16x4_f32(
                /*neg_a=*/false, a, /*neg_b=*/false, b2,
                /*c_mod=*/(short)0, c, /*reuse_a=*/false, /*reuse_b=*/false);
    }

    // Epilogue: bias add, store raw h, per-(wave,half) column partials.
    const int   col = nBase + l15;
    const float bb  = bias[col];
    float s = 0.f, q = 0.f;
    #pragma unroll
    for (int i = 0; i < 8; ++i) {
        float v = c[i] + bb;
        const int row = mBase + half * 8 + i;
        h[(size_t)row * E_DIM + col] = v;
        s += v;
        q += v * v;
    }

    __shared__ float sh_s[256];
    __shared__ float sh_q[256];
    sh_s[threadIdx.x] = s;
    sh_q[threadIdx.x] = q;
    __syncthreads();

    // Column `t` lives in wave t/16, lanes (t%16) and (t%16)+16.
    if (threadIdx.x < E_DIM) {
        const int w   = threadIdx.x >> 4;
        const int c15 = threadIdx.x & 15;
        const float S = sh_s[w * 32 + c15] + sh_s[w * 32 + 16 + c15];
        const float Q = sh_q[w * 32 + c15] + sh_q[w * 32 + 16 + c15];
        psum[(size_t)blockIdx.x * E_DIM + threadIdx.x] = S;
        psq [(size_t)blockIdx.x * E_DIM + threadIdx.x] = Q;
    }
}

// ---------------------------------------------------------------------------
// Kernel 2: reduce 3125 partials per column -> mu, rstd. 1 block, 128 threads.
// ---------------------------------------------------------------------------
__global__ __launch_bounds__(E_DIM)
void stats_final(const float* __restrict__ psum, const float* __restrict__ psq,
                 float* __restrict__ mu, float* __restrict__ rstd, int nblk)
{
    const int c = threadIdx.x;
    float S = 0.f, Q = 0.f;
    for (int b = 0; b < nblk; ++b) {
        S += psum[(size_t)b * E_DIM + c];
        Q += psq [(size_t)b * E_DIM + c];
    }
    const float m   = S / (float)U_ROWS;
    const float var = Q / (float)U_ROWS - m * m;   // biased, like BatchNorm1d
    mu[c]   = m;
    rstd[c] = rsqrtf(var + BN_EPS);
}

// ---------------------------------------------------------------------------
// Kernel 3: one block (128 threads) per output row. Dedup the 17 indices
// (duplicates collapse under mask.set(1.0)), then mean of tanh(BN(h[idx])).
// ---------------------------------------------------------------------------
__global__ __launch_bounds__(E_DIM)
void aggregate(const float* __restrict__ h, const int* __restrict__ col_idx,
               const float* __restrict__ mu, const float* __restrict__ rstd,
               const float* __restrict__ gamma, const float* __restrict__ beta,
               float* __restrict__ out)
{
    __shared__ int   s_idx[NSAMP];
    __shared__ int   s_valid[NSAMP];
    __shared__ float s_inv;

    const int row = blockIdx.x;
    const int t   = threadIdx.x;

    if (t < NSAMP) s_idx[t] = col_idx[row * NSAMP + t];
    __syncthreads();

    if (t < NSAMP) {
        const int mine = s_idx[t];
        int v = 1;
        for (int j = 0; j < t; ++j)
            if (s_idx[j] == mine) { v = 0; break; }
        s_valid[t] = v;
    }
    __syncthreads();

    if (t == 0) {
        int cnt = 0;
        for (int j = 0; j < NSAMP; ++j) cnt += s_valid[j];
        s_inv = 1.0f / (float)cnt;
    }
    __syncthreads();

    const float m  = mu[t];
    const float r  = rstd[t];
    const float g  = gamma[t];
    const float be = beta[t];

    float acc = 0.f;
    #pragma unroll
    for (int i = 0; i < NSAMP; ++i) {
        if (s_valid[i]) {
            const float v = h[(size_t)s_idx[i] * E_DIM + t];
            acc += tanhf((v - m) * r * g + be);
        }
    }
    out[(size_t)row * E_DIM + t] = acc * s_inv;
}

// ---------------------------------------------------------------------------
// Host-side launcher (graph-capture safe: stream launches only).
// Inputs: 0=features 1=W 2=b 3=gamma 4=beta 5=row_idx 6=col_idx 7=B
// ---------------------------------------------------------------------------
extern "C" void kernel_launch(void* const* d_in, const int* in_sizes, int n_in,
                              void* d_out, int out_size, void* d_ws, size_t ws_size,
                              hipStream_t stream)
{
    (void)in_sizes; (void)n_in; (void)ws_size;

    const float* feat    = (const float*)d_in[0];
    const float* W       = (const float*)d_in[1];
    const float* bias    = (const float*)d_in[2];
    const float* gamma   = (const float*)d_in[3];
    const float* beta    = (const float*)d_in[4];
    const int*   col_idx = (const int*)  d_in[6];   // row_idx[5] implied by layout
    float*       out     = (float*)d_out;

    const int B    = out_size / E_DIM;   // 4096
    const int nblk = U_ROWS / 16;        // 3125 (exact)

    // Workspace layout (fp32): h[U*E] | psum[nblk*E] | psq[nblk*E] | mu[E] | rstd[E]
    float* h    = (float*)d_ws;
    float* psum = h    + (size_t)U_ROWS * E_DIM;
    float* psq  = psum + (size_t)nblk   * E_DIM;
    float* mu   = psq  + (size_t)nblk   * E_DIM;
    float* rstd = mu   + E_DIM;

    gemm_bias_stats<<<nblk, 256, 0, stream>>>(feat, W, bias, h, psum, psq);
    stats_final   <<<1,    E_DIM, 0, stream>>>(psum, psq, mu, rstd, nblk);
    aggregate     <<<B,    E_DIM, 0, stream>>>(h, col_idx, mu, rstd, gamma, beta, out);
}